// GAT_609885356262
// MI455X (gfx1250) — compile-verified
//
#include <hip/hip_runtime.h>
#include <hip/hip_bf16.h>

// ---------------- model dims ----------------
#define V      1000
#define BATCH  32
#define NFEAT  64
#define NHID   64
#define NHEADS 8
#define NCLASS 10
#define POOL   8
#define NN_EMBED 64
#define EMBED  256
#define ALPHA  0.2f
#define MASKVAL (-9.0e15f)

typedef float v2f __attribute__((ext_vector_type(2)));
typedef float v8f __attribute__((ext_vector_type(8)));

// ---------------- workspace layout (floats) ----------------
constexpr long OFF_L    = 0;                        // V*V
constexpr long OFF_LMAX = OFF_L + (long)V * V;      // 16
constexpr long OFF_DINV = OFF_LMAX + 16;            // 1024
constexpr long OFF_X0   = OFF_DINV + 1024;          // V*BATCH
constexpr long OFF_X1   = OFF_X0 + (long)V * BATCH;
constexpr long OFF_X2   = OFF_X1 + (long)V * BATCH;
constexpr long OFF_H    = OFF_X2 + (long)V * BATCH;            // BATCH*V*NFEAT
constexpr long OFF_HW   = OFF_H + (long)BATCH * V * NFEAT;     // BATCH*V*NHID
constexpr long OFF_F1   = OFF_HW + (long)BATCH * V * NHID;     // BATCH*V
constexpr long OFF_F2   = OFF_F1 + (long)BATCH * V;
constexpr long OFF_M    = OFF_F2 + (long)BATCH * V;
constexpr long OFF_S    = OFF_M + (long)BATCH * V;
constexpr long OFF_HO   = OFF_S + (long)BATCH * V;             // BATCH*V*NHID
constexpr long OFF_POOL = OFF_HO + (long)BATCH * V * NHID;     // BATCH*64000
constexpr long OFF_HFC  = OFF_POOL + (long)BATCH * (V / POOL) * NHID * NHEADS; // BATCH*EMBED
constexpr long OFF_Z1   = OFF_HFC + (long)BATCH * EMBED;       // BATCH*512
constexpr long OFF_Z2   = OFF_Z1 + (long)BATCH * 512;          // BATCH*NN_EMBED

// ============================================================
// 1) degree -> dinv = rsqrt(rowsum)
// ============================================================
__global__ void gatk_degree(const float* __restrict__ adj, float* __restrict__ dinv) {
    int v = blockIdx.x * blockDim.x + threadIdx.x;
    if (v >= V) return;
    const float* row = adj + (long)v * V;
    float d = 0.f;
    for (int j = 0; j < V; ++j) d += row[j];
    dinv[v] = d > 0.f ? rsqrtf(d) : 0.f;
}

// ============================================================
// 2) L = I - D^-1/2 A D^-1/2
// ============================================================
__global__ void gatk_build_L(const float* __restrict__ adj, const float* __restrict__ dinv,
                             float* __restrict__ L) {
    long idx = (long)blockIdx.x * blockDim.x + threadIdx.x;
    if (idx >= (long)V * V) return;
    int i = (int)(idx / V), j = (int)(idx % V);
    L[idx] = (i == j ? 1.f : 0.f) - dinv[i] * adj[idx] * dinv[j];
}

// ============================================================
// 3) power iteration for lmax (single workgroup, LDS vector)
// ============================================================
__global__ __launch_bounds__(1024)
void gatk_power_iter(const float* __restrict__ L, float* __restrict__ lmax_out) {
    __shared__ float sv[V];
    __shared__ float red[32];
    int t = threadIdx.x;
    for (int v = t; v < V; v += 1024) sv[v] = (v & 1) ? -1.f : 1.f;  // high-freq init
    __syncthreads();
    float lam = 1.f;
    for (int it = 0; it < 64; ++it) {
        float w = 0.f;
        if (t < V) {
            const float* row = L + (long)t * V;
            for (int j = 0; j < V; ++j) w += row[j] * sv[j];
        }
        float ss = w * w;
        for (int off = 16; off > 0; off >>= 1) ss += __shfl_xor(ss, off, 32);
        if ((t & 31) == 0) red[t >> 5] = ss;
        __syncthreads();
        if (t < 32) {
            float v2 = red[t];
            for (int off = 16; off > 0; off >>= 1) v2 += __shfl_xor(v2, off, 32);
            if (t == 0) red[0] = v2;
        }
        __syncthreads();
        float norm = sqrtf(red[0]);
        lam = norm;
        float inv = norm > 0.f ? 1.f / norm : 0.f;
        if (t < V) sv[t] = w * inv;
        __syncthreads();
    }
    if (t == 0) lmax_out[0] = lam;
}

// ============================================================
// 4) Ls = 2*L/lmax - I  (in place)
// ============================================================
__global__ void gatk_rescale(float* __restrict__ L, const float* __restrict__ lmax) {
    long idx = (long)blockIdx.x * blockDim.x + threadIdx.x;
    if (idx >= (long)V * V) return;
    int i = (int)(idx / V), j = (int)(idx % V);
    L[idx] = (2.0f / lmax[0]) * L[idx] - (i == j ? 1.f : 0.f);
}

// ============================================================
// 5) x0 = x.T  ([B,V] -> [V,B])
// ============================================================
__global__ void gatk_transpose_x(const float* __restrict__ x, float* __restrict__ x0) {
    int idx = blockIdx.x * blockDim.x + threadIdx.x;
    if (idx >= BATCH * V) return;
    int v = idx % V, b = idx / V;
    x0[(long)v * BATCH + b] = x[idx];
}

// ============================================================
// shared GEMM epilogue: predicated stores AFTER the WMMA loop
// ============================================================
__device__ __forceinline__
void gemm_epilogue(v8f acc, int m0, int n0, int half, int lm,
                   const float* bias, const float* Dsrc, float* C,
                   int M, int N, float alpha, float beta, int relu) {
    int cn = n0 + lm;
    if (cn < N) {
        #pragma unroll
        for (int r = 0; r < 8; ++r) {
            int cm = m0 + r + 8 * half;
            if (cm < M) {
                float vv = alpha * acc[r];
                if (Dsrc) vv += beta * Dsrc[(long)cm * N + cn];
                if (bias) vv += bias[cn];
                if (relu) vv = fmaxf(vv, 0.f);
                C[(long)cm * N + cn] = vv;
            }
        }
    }
}

// ============================================================
// fp32 WMMA GEMM, B not transposed: C[M,N] = alpha*A[M,K]@B[K,N] (+beta*Dsrc+bias)
//   one wave per 16x16 tile; branch-free k-loop, pointer-increment addressing.
// ============================================================
__global__ void gatk_gemm_nn(const float* __restrict__ A, const float* __restrict__ Bm,
                             const float* __restrict__ bias, const float* __restrict__ Dsrc,
                             float* __restrict__ C,
                             int M, int N, int K,
                             float alpha, float beta, int relu) {
    int lane = threadIdx.x;
    int half = lane >> 4;          // 0: K{0,1}, 1: K{2,3}
    int lm   = lane & 15;
    int m0 = blockIdx.x * 16;
    int n0 = blockIdx.y * 16;
    int am = min(m0 + lm, M - 1);
    int bn = min(n0 + lm, N - 1);
    const float* Ap = A + (long)am * K + 2 * half;          // advances by 4
    const float* Bp = Bm + (long)(2 * half) * N + bn;       // advances by 4*N
    v8f acc = {};
    #pragma unroll 4
    for (int k = 0; k < K; k += 4) {
        v2f a = *(const v2f*)Ap;           // contiguous pair -> b64 load
        v2f b; b.x = Bp[0]; b.y = Bp[N];   // two rows, stride N
        acc = __builtin_amdgcn_wmma_f32_16x16x4_f32(false, a, false, b, (short)0, acc,
                                                    false, false);
        Ap += 4;
        Bp += 4 * (long)N;
    }
    gemm_epilogue(acc, m0, n0, half, lm, bias, Dsrc, C, M, N, alpha, beta, relu);
}

// ============================================================
// fp32 WMMA GEMM, B transposed: C[M,N] = alpha*A[M,K]@Bm[N,K]^T (+beta*Dsrc+bias)
//   contiguous b64 loads for both fragments; stream prefetch on A.
// ============================================================
__global__ void gatk_gemm_tn(const float* __restrict__ A, const float* __restrict__ Bm,
                             const float* __restrict__ bias, const float* __restrict__ Dsrc,
                             float* __restrict__ C,
                             int M, int N, int K,
                             float alpha, float beta, int relu) {
    int lane = threadIdx.x;
    int half = lane >> 4;
    int lm   = lane & 15;
    int m0 = blockIdx.x * 16;
    int n0 = blockIdx.y * 16;
    int am = min(m0 + lm, M - 1);
    int bn = min(n0 + lm, N - 1);
    const float* Ap = A + (long)am * K + 2 * half;          // advances by 4
    const float* Bp = Bm + (long)bn * K + 2 * half;         // advances by 4
    v8f acc = {};
    int k = 0;
    for (; k + 16 <= K; k += 16) {                          // main: 4 WMMA per block
        __builtin_prefetch(Ap + 256, 0, 1);                 // unconditional, no branch
        __builtin_prefetch(Bp + 256, 0, 1);
        #pragma unroll
        for (int u = 0; u < 4; ++u) {
            v2f a = *(const v2f*)(Ap + 4 * u);
            v2f b = *(const v2f*)(Bp + 4 * u);
            acc = __builtin_amdgcn_wmma_f32_16x16x4_f32(false, a, false, b, (short)0, acc,
                                                        false, false);
        }
        Ap += 16;
        Bp += 16;
    }
    for (; k < K; k += 4) {                                 // remainder (K % 16)
        v2f a = *(const v2f*)Ap;
        v2f b = *(const v2f*)Bp;
        acc = __builtin_amdgcn_wmma_f32_16x16x4_f32(false, a, false, b, (short)0, acc,
                                                    false, false);
        Ap += 4;
        Bp += 4;
    }
    gemm_epilogue(acc, m0, n0, half, lm, bias, Dsrc, C, M, N, alpha, beta, relu);
}

// ============================================================
// 6) cheby combine: h[b,v,f] = relu(sum_k Xk[k][v][b]*cl_w[f,k] + cl_b[f])
// ============================================================
__global__ void gatk_cheby_combine(const float* __restrict__ x0, const float* __restrict__ x1,
                                   const float* __restrict__ x2, const float* __restrict__ cl_w,
                                   const float* __restrict__ cl_b, float* __restrict__ h) {
    long idx = (long)blockIdx.x * blockDim.x + threadIdx.x;
    if (idx >= (long)BATCH * V * NFEAT) return;
    int f = (int)(idx & 63);
    long t = idx >> 6;
    int v = (int)(t % V), b = (int)(t / V);
    long vb = (long)v * BATCH + b;
    float val = cl_b[f] + x0[vb] * cl_w[f * 3 + 0] + x1[vb] * cl_w[f * 3 + 1]
              + x2[vb] * cl_w[f * 3 + 2];
    h[idx] = fmaxf(val, 0.f);
}

// ============================================================
// 7) f1/f2 per (b,v): projections of hW onto attention vector halves
// ============================================================
__global__ void gatk_f12(const float* __restrict__ hW, const float* __restrict__ att_a,
                         int head, float* __restrict__ f1, float* __restrict__ f2) {
    int idx = blockIdx.x * blockDim.x + threadIdx.x;   // b*V + v
    if (idx >= BATCH * V) return;
    const float* hr = hW + (long)idx * NHID;
    const float* a1 = att_a + head * 2 * NHID;
    const float* a2 = a1 + NHID;
    float s1 = 0.f, s2 = 0.f;
    #pragma unroll 8
    for (int f = 0; f < NHID; ++f) { float hv = hr[f]; s1 += hv * a1[f]; s2 += hv * a2[f]; }
    f1[idx] = s1; f2[idx] = s2;
}

// ============================================================
// 8) softmax row stats over masked leaky-relu logits
// ============================================================
__global__ void gatk_rowstats(const float* __restrict__ adj, const float* __restrict__ f1,
                              const float* __restrict__ f2, float* __restrict__ mrow,
                              float* __restrict__ srow) {
    int i = blockIdx.x, b = blockIdx.y, lane = threadIdx.x;
    float f1i = f1[b * V + i];
    const float* adjr = adj + (long)i * V;
    const float* f2b  = f2 + b * V;
    float mx = -3.0e38f;
    for (int j = lane; j < V; j += 32) {
        float e = f1i + f2b[j]; e = e > 0.f ? e : ALPHA * e;
        e = adjr[j] > 0.f ? e : MASKVAL;
        mx = fmaxf(mx, e);
    }
    for (int off = 16; off > 0; off >>= 1) mx = fmaxf(mx, __shfl_xor(mx, off, 32));
    float s = 0.f;
    for (int j = lane; j < V; j += 32) {
        float e = f1i + f2b[j]; e = e > 0.f ? e : ALPHA * e;
        e = adjr[j] > 0.f ? e : MASKVAL;
        s += __expf(e - mx);
    }
    for (int off = 16; off > 0; off >>= 1) s += __shfl_xor(s, off, 32);
    if (lane == 0) { mrow[b * V + i] = mx; srow[b * V + i] = s; }
}

// ============================================================
// 9) flash-style attention apply: out[b,i,:] = relu(sum_j p(b,i,j) * hW[b,j,:])
//    one wave per (16-row i-tile, batch); A-frag = P tile built on the fly,
//    4 WMMA accumulators cover the 64 output features. (elu∘relu == relu)
// ============================================================
__global__ void gatk_attn_apply(const float* __restrict__ adj, const float* __restrict__ hW,
                                const float* __restrict__ f1, const float* __restrict__ f2,
                                const float* __restrict__ mrow, const float* __restrict__ srow,
                                float* __restrict__ out) {
    int itile = blockIdx.x;
    int b     = blockIdx.y;
    int lane  = threadIdx.x;
    int half  = lane >> 4, lm = lane & 15;
    int i = itile * 16 + lm;
    int ic = min(i, V - 1);
    float pscale = (i < V) ? (1.0f / srow[b * V + ic]) : 0.0f;
    float f1i = f1[b * V + ic];
    float mi  = mrow[b * V + ic];
    // per-lane streaming pointers (advance by constant strides; no idx math in loop)
    const float* pf2 = f2 + b * V + 2 * half;                    // += 4
    const float* pad = adj + (long)ic * V + 2 * half;            // += 4
    const float* ph  = hW + ((long)b * V + 2 * half) * NHID;     // += 4*NHID
    v8f acc0 = {}, acc1 = {}, acc2 = {}, acc3 = {};
    #pragma unroll 4
    for (int j = 0; j < V; j += 4) {
        v2f fp = *(const v2f*)pf2;         // f2[k0], f2[k1] contiguous
        v2f ap = *(const v2f*)pad;         // adj[i,k0], adj[i,k1] contiguous
        float e0 = f1i + fp.x; e0 = e0 > 0.f ? e0 : ALPHA * e0;
        float e1 = f1i + fp.y; e1 = e1 > 0.f ? e1 : ALPHA * e1;
        e0 = ap.x > 0.f ? e0 : MASKVAL;
        e1 = ap.y > 0.f ? e1 : MASKVAL;
        v2f a;
        a.x = __expf(e0 - mi) * pscale;
        a.y = __expf(e1 - mi) * pscale;
        v2f b0; b0.x = ph[lm];           b0.y = ph[NHID + lm];
        v2f b1; b1.x = ph[16 + lm];      b1.y = ph[NHID + 16 + lm];
        v2f b2; b2.x = ph[32 + lm];      b2.y = ph[NHID + 32 + lm];
        v2f b3; b3.x = ph[48 + lm];      b3.y = ph[NHID + 48 + lm];
        acc0 = __builtin_amdgcn_wmma_f32_16x16x4_f32(false, a, false, b0, (short)0, acc0, false, false);
        acc1 = __builtin_amdgcn_wmma_f32_16x16x4_f32(false, a, false, b1, (short)0, acc1, false, false);
        acc2 = __builtin_amdgcn_wmma_f32_16x16x4_f32(false, a, false, b2, (short)0, acc2, false, false);
        acc3 = __builtin_amdgcn_wmma_f32_16x16x4_f32(false, a, false, b3, (short)0, acc3, false, false);
        pf2 += 4;
        pad += 4;
        ph  += 4 * NHID;
    }
    #pragma unroll
    for (int r = 0; r < 8; ++r) {
        int m = itile * 16 + r + 8 * half;
        if (m < V) {
            float* orow = out + ((long)b * V + m) * NHID;
            orow[lm]      = fmaxf(acc0[r], 0.f);
            orow[16 + lm] = fmaxf(acc1[r], 0.f);
            orow[32 + lm] = fmaxf(acc2[r], 0.f);
            orow[48 + lm] = fmaxf(acc3[r], 0.f);
        }
    }
}

// ============================================================
// 10) pool over POOL=8 nodes into the concat layout [B, 125*512]
// ============================================================
__global__ void gatk_pool(const float* __restrict__ ho, float* __restrict__ pooled, int head) {
    int idx = blockIdx.x * blockDim.x + threadIdx.x;   // b*(125*64) + vp*64 + f
    if (idx >= BATCH * (V / POOL) * NHID) return;
    int f  = idx & 63;
    int t  = idx >> 6;
    int vp = t % (V / POOL);
    int b  = t / (V / POOL);
    const float* base = ho + ((long)b * V + vp * POOL) * NHID + f;
    float mx = base[0];
    #pragma unroll
    for (int r = 1; r < POOL; ++r) mx = fmaxf(mx, base[(long)r * NHID]);
    pooled[(long)b * ((V / POOL) * NHID * NHEADS) + vp * (NHID * NHEADS) + head * NHID + f] = mx;
}

// ============================================================
// 11) fc2 + relu + log_softmax over concat(hfc, z2)
// ============================================================
__global__ void gatk_head(const float* __restrict__ hfc, const float* __restrict__ z2,
                          const float* __restrict__ fc2_w, const float* __restrict__ fc2_b,
                          float* __restrict__ out) {
    int b = blockIdx.x, lane = threadIdx.x;
    float o = -3.0e38f, logit = 0.f;
    if (lane < NCLASS) {
        const float* wrow = fc2_w + lane * (EMBED + NN_EMBED);
        float s = fc2_b[lane];
        const float* hb = hfc + (long)b * EMBED;
        for (int q = 0; q < EMBED; ++q) s += hb[q] * wrow[q];
        const float* zb = z2 + (long)b * NN_EMBED;
        for (int q = 0; q < NN_EMBED; ++q) s += zb[q] * wrow[EMBED + q];
        logit = fmaxf(s, 0.f);
        o = logit;
    }
    float mx = o;
    for (int off = 16; off > 0; off >>= 1) mx = fmaxf(mx, __shfl_xor(mx, off, 32));
    float e = lane < NCLASS ? __expf(logit - mx) : 0.f;
    float se = e;
    for (int off = 16; off > 0; off >>= 1) se += __shfl_xor(se, off, 32);
    if (lane < NCLASS) out[(long)BATCH * V + b * NCLASS + lane] = logit - mx - logf(se);
}

// ============================================================
// launch
// ============================================================
extern "C" void kernel_launch(void* const* d_in, const int* in_sizes, int n_in,
                              void* d_out, int out_size, void* d_ws, size_t ws_size,
                              hipStream_t stream) {
    const float* x     = (const float*)d_in[0];
    const float* adj   = (const float*)d_in[1];
    /* d_in[2] = conv_degree (int scalar, fixed 3) */
    const float* cl_w  = (const float*)d_in[3];
    const float* cl_b  = (const float*)d_in[4];
    const float* att_W = (const float*)d_in[5];
    const float* att_a = (const float*)d_in[6];
    const float* fc1_w = (const float*)d_in[7];
    const float* fc1_b = (const float*)d_in[8];
    const float* fc2_w = (const float*)d_in[9];
    const float* fc2_b = (const float*)d_in[10];
    const float* dec_w = (const float*)d_in[11];
    const float* dec_b = (const float*)d_in[12];
    const float* nn1_w = (const float*)d_in[13];
    const float* nn1_b = (const float*)d_in[14];
    const float* nn2_w = (const float*)d_in[15];
    const float* nn2_b = (const float*)d_in[16];
    float* out = (float*)d_out;
    float* ws  = (float*)d_ws;

    float* L     = ws + OFF_L;
    float* lmax  = ws + OFF_LMAX;
    float* dinv  = ws + OFF_DINV;
    float* x0    = ws + OFF_X0;
    float* x1    = ws + OFF_X1;
    float* x2    = ws + OFF_X2;
    float* h     = ws + OFF_H;
    float* hWbuf = ws + OFF_HW;
    float* f1    = ws + OFF_F1;
    float* f2    = ws + OFF_F2;
    float* mrow  = ws + OFF_M;
    float* srow  = ws + OFF_S;
    float* ho    = ws + OFF_HO;
    float* pooled= ws + OFF_POOL;
    float* hfc   = ws + OFF_HFC;
    float* z1    = ws + OFF_Z1;
    float* z2    = ws + OFF_Z2;

    // Laplacian + lmax + rescale
    gatk_degree<<<(V + 255) / 256, 256, 0, stream>>>(adj, dinv);
    gatk_build_L<<<(int)(((long)V * V + 255) / 256), 256, 0, stream>>>(adj, dinv, L);
    gatk_power_iter<<<1, 1024, 0, stream>>>(L, lmax);
    gatk_rescale<<<(int)(((long)V * V + 255) / 256), 256, 0, stream>>>(L, lmax);

    // Chebyshev: x0 = x.T ; x1 = Ls@x0 ; x2 = 2*Ls@x1 - x0
    gatk_transpose_x<<<(BATCH * V + 255) / 256, 256, 0, stream>>>(x, x0);
    gatk_gemm_nn<<<dim3((V + 15) / 16, BATCH / 16), 32, 0, stream>>>(
        L, x0, nullptr, nullptr, x1, V, BATCH, V, 1.0f, 0.0f, 0);
    gatk_gemm_nn<<<dim3((V + 15) / 16, BATCH / 16), 32, 0, stream>>>(
        L, x1, nullptr, x0, x2, V, BATCH, V, 2.0f, -1.0f, 0);
    gatk_cheby_combine<<<(int)(((long)BATCH * V * NFEAT + 255) / 256), 256, 0, stream>>>(
        x0, x1, x2, cl_w, cl_b, h);

    // GAT heads (sequential, buffers reused)
    for (int hd = 0; hd < NHEADS; ++hd) {
        gatk_gemm_nn<<<dim3(BATCH * V / 16, NHID / 16), 32, 0, stream>>>(
            h, att_W + (long)hd * NFEAT * NHID, nullptr, nullptr, hWbuf,
            BATCH * V, NHID, NFEAT, 1.0f, 0.0f, 0);
        gatk_f12<<<(BATCH * V + 255) / 256, 256, 0, stream>>>(hWbuf, att_a, hd, f1, f2);
        gatk_rowstats<<<dim3(V, BATCH), 32, 0, stream>>>(adj, f1, f2, mrow, srow);
        gatk_attn_apply<<<dim3((V + 15) / 16, BATCH), 32, 0, stream>>>(
            adj, hWbuf, f1, f2, mrow, srow, ho);
        gatk_pool<<<(BATCH * (V / POOL) * NHID + 255) / 256, 256, 0, stream>>>(ho, pooled, hd);
    }

    // fc1: hfc = relu(pooled @ fc1_w^T + fc1_b)   [32 x 64000] x [256 x 64000]^T
    gatk_gemm_tn<<<dim3(BATCH / 16, EMBED / 16), 32, 0, stream>>>(
        pooled, fc1_w, fc1_b, nullptr, hfc,
        BATCH, EMBED, (V / POOL) * NHID * NHEADS, 1.0f, 0.0f, 1);

    // decoder: x_rec = hfc @ dec_w^T + dec_b  -> out[0 .. 32*1000)
    gatk_gemm_tn<<<dim3(BATCH / 16, (V + 15) / 16), 32, 0, stream>>>(
        hfc, dec_w, dec_b, nullptr, out, BATCH, V, EMBED, 1.0f, 0.0f, 0);

    // nn branch: z1 = relu(x @ nn1_w^T + b); z2 = relu(z1 @ nn2_w^T + b)
    gatk_gemm_tn<<<dim3(BATCH / 16, 512 / 16), 32, 0, stream>>>(
        x, nn1_w, nn1_b, nullptr, z1, BATCH, 512, V, 1.0f, 0.0f, 1);
    gatk_gemm_tn<<<dim3(BATCH / 16, NN_EMBED / 16), 32, 0, stream>>>(
        z1, nn2_w, nn2_b, nullptr, z2, BATCH, NN_EMBED, 512, 1.0f, 0.0f, 1);

    // fc2 + relu + log_softmax -> out[32000 .. 32320)
    gatk_head<<<BATCH, 32, 0, stream>>>(hfc, z2, fc2_w, fc2_b, out);
}